// BackwardStep_38749194944853
// MI455X (gfx1250) — compile-verified
//
#include <hip/hip_runtime.h>

// CDNA5 / gfx1250: wave32, WMMA f32 16x16x4, 320KB LDS per WGP.
typedef __attribute__((ext_vector_type(2))) float v2f;
typedef __attribute__((ext_vector_type(8))) float v8f;

#define ALPHA  1.0f
#define EPS    1e-4f
#define RHO    0.1f
#define SIGMA  1e-6f
#define NITER  100

#define NX     128
#define MI     128
#define ME     32
#define MTOT   192          // mi + 2*me
#define A_W    129          // padded LDS row stride for A   (odd -> conflict-free columns)
#define AUG_W  257          // padded stride for [K | Kinv]  (odd -> conflict-free rows+cols)
#define D_W    17           // padded stride for panel multiplier block D

// LDS layout (floats):
//   sA   : 192*129 = 24768
//   aug  : 128*257 = 32896
//   Pold : 16*257  =  4112   (saved panel rows for blocked Gauss-Jordan)
//   Dm   : 128*17  =  2176   (panel multiplier block)
//   sS,sI: 2*16*17 =   544   (16x16 diag block + its inverse)
//   vectors        =  1152
#define SMEM_FLOATS (MTOT*A_W + NX*AUG_W + 16*AUG_W + NX*D_W + 2*16*17 + 1152)
#define SMEM_BYTES  (SMEM_FLOATS * 4)

__global__ __launch_bounds__(256) void admm_qp_kernel(
    const float* __restrict__ x_in,   // [N,128,1]
    const float* __restrict__ Q,      // [N,128,128]
    const float* __restrict__ q,      // [N,128,1]
    const float* __restrict__ Ain,    // [N,128,128]
    const float* __restrict__ bin,    // [N,128,1]
    const float* __restrict__ Aeq,    // [N,32,128]
    const float* __restrict__ beq,    // [N,32,1]
    float* __restrict__ out)          // [N,128,1]
{
    extern __shared__ __align__(16) char smem_raw[];
    float* sA   = (float*)smem_raw;           // 192 x 129
    float* aug  = sA   + MTOT * A_W;          // 128 x 257
    float* Pold = aug  + NX * AUG_W;          // 16 x 257
    float* Dm   = Pold + 16 * AUG_W;          // 128 x 17
    float* sS   = Dm   + NX * D_W;            // 16 x 17
    float* sI   = sS   + 16 * 17;             // 16 x 17
    float* q2   = sI   + 16 * 17;             // 128
    float* uu   = q2   + NX;                  // 192
    float* xv   = uu   + MTOT;                // 128
    float* zv   = xv   + NX;                  // 192
    float* yv   = zv   + MTOT;                // 192
    float* vv   = yv   + MTOT;                // 192
    float* rhs  = vv   + MTOT;                // 128

    const int b   = blockIdx.x;
    const int tid = threadIdx.x;
    const int wave = tid >> 5;
    const int lane = tid & 31;
    const int n    = lane & 15;   // fragment column / A-row index (M == n)
    const int hi   = lane >> 4;   // half-wave select
    const int kb   = hi << 1;     // K base within 4-wide slice: 0 or 2
    const int rr   = tid >> 4;    // 16x16 element ops: row
    const int cc   = tid & 15;    //                    col

    const float* Qb = Q   + (size_t)b * NX * NX;
    const float* Ai = Ain + (size_t)b * MI * NX;
    const float* Ae = Aeq + (size_t)b * ME * NX;

    // ---------- Phase 0: stage A = [A_ineq; A_eq; -A_eq], u, q2 into LDS ----------
    for (int idx = tid; idx < MTOT * NX; idx += 256) {
        int m = idx >> 7;
        int i = idx & (NX - 1);
        float v;
        if (m < MI)            v =  Ai[m * NX + i];
        else if (m < MI + ME)  v =  Ae[(m - MI) * NX + i];
        else                   v = -Ae[(m - MI - ME) * NX + i];
        sA[m * A_W + i] = v;
    }
    if (tid < NX) {
        q2[tid] = q[(size_t)b * NX + tid] - ALPHA * x_in[(size_t)b * NX + tid];
        xv[tid] = 0.0f;
    }
    if (tid < MTOT) {
        float u;
        if (tid < MI)           u =  bin[(size_t)b * MI + tid];
        else if (tid < MI + ME) u =  beq[(size_t)b * ME + (tid - MI)] + EPS;
        else                    u = -beq[(size_t)b * ME + (tid - MI - ME)];
        uu[tid] = u;
        zv[tid] = 0.0f;
        yv[tid] = 0.0f;
    }
    // overlap Q fetch with the SYRK below (global_prefetch_b8)
    if (tid < NX) __builtin_prefetch(Qb + tid * NX, 0, 0);
    __syncthreads();

    // ---------- Phase 1: K = Q + (ALPHA+SIGMA) I + RHO * A^T A  (WMMA f32 16x16x4) ----------
    for (int t = wave; t < 64; t += 8) {
        const int TI = t >> 3;
        const int TJ = t & 7;
        const int ci = TI * 16 + n;      // A^T row  == column of A
        const int cj = TJ * 16 + n;      // B column
        v8f acc = {0.f, 0.f, 0.f, 0.f, 0.f, 0.f, 0.f, 0.f};
        #pragma unroll 4
        for (int m0 = 0; m0 < MTOT; m0 += 4) {
            v2f af, bf;
            af.x = sA[(m0 + kb    ) * A_W + ci];
            af.y = sA[(m0 + kb + 1) * A_W + ci];
            bf.x = sA[(m0 + kb    ) * A_W + cj];
            bf.y = sA[(m0 + kb + 1) * A_W + cj];
            acc = __builtin_amdgcn_wmma_f32_16x16x4_f32(
                    false, af, false, bf, (short)0, acc, false, false);
        }
        #pragma unroll
        for (int v = 0; v < 8; ++v) {
            int row = TI * 16 + v + 8 * hi;
            int col = TJ * 16 + n;
            float kval = RHO * acc[v] + Qb[row * NX + col];
            if (row == col) kval += (ALPHA + SIGMA);
            aug[row * AUG_W + col] = kval;
        }
    }
    // identity in right half of augmented matrix
    for (int idx = tid; idx < NX * NX; idx += 256) {
        int r = idx >> 7;
        int c = idx & (NX - 1);
        aug[r * AUG_W + NX + c] = (r == c) ? 1.0f : 0.0f;
    }
    __syncthreads();

    // ---------- Phase 2: blocked Gauss-Jordan inverse, rank-16 updates on WMMA ----------
    // Per 16-col panel: A <- (I + D*Sel)*A with
    //   D[r,:]      = -A[r,panel] * inv(App)   (r not in panel)
    //   D[panel,:]  =  inv(App) - I
    // so  A += D * Pold, where Pold = saved panel rows. K is SPD -> no pivoting.
    for (int p0 = 0; p0 < NX; p0 += 16) {
        // (a) save panel rows; copy App; init identity
        for (int idx = tid; idx < 16 * 256; idx += 256) {
            int r = idx >> 8, c = idx & 255;
            Pold[r * AUG_W + c] = aug[(p0 + r) * AUG_W + c];
        }
        sS[rr * 17 + cc] = aug[(p0 + rr) * AUG_W + p0 + cc];
        sI[rr * 17 + cc] = (rr == cc) ? 1.0f : 0.0f;
        __syncthreads();

        // (b) invert 16x16 App: one element per thread, unpivoted GJ
        for (int k = 0; k < 16; ++k) {
            float pivinv = 1.0f / sS[k * 17 + k];
            float fr = sS[rr * 17 + k] * pivinv;   // multiplier for this row
            float ps = sS[k * 17 + cc];            // pivot row (unscaled)
            float pi = sI[k * 17 + cc];
            __syncthreads();
            if (rr == k) {
                sS[rr * 17 + cc] = ps * pivinv;
                sI[rr * 17 + cc] = pi * pivinv;
            } else {
                sS[rr * 17 + cc] -= fr * ps;
                sI[rr * 17 + cc] -= fr * pi;
            }
            __syncthreads();
        }
        // sI == inv(App)

        // (c) D = -aug[:,panel] * inv(App)  — one 16x16 tile per wave (WMMA)
        {
            const int TI = wave;
            v8f acc = {0.f, 0.f, 0.f, 0.f, 0.f, 0.f, 0.f, 0.f};
            #pragma unroll
            for (int k0 = 0; k0 < 16; k0 += 4) {
                v2f af, bf;
                af.x =  aug[(TI * 16 + n) * AUG_W + p0 + k0 + kb    ];
                af.y =  aug[(TI * 16 + n) * AUG_W + p0 + k0 + kb + 1];
                bf.x = -sI[(k0 + kb    ) * 17 + n];   // fold the minus here:
                bf.y = -sI[(k0 + kb + 1) * 17 + n];   // f32 WMMA NEG only allows CNeg
                acc = __builtin_amdgcn_wmma_f32_16x16x4_f32(
                        false, af, false, bf, (short)0, acc, false, false);
            }
            #pragma unroll
            for (int v = 0; v < 8; ++v)
                Dm[(TI * 16 + v + 8 * hi) * D_W + n] = acc[v];
        }
        __syncthreads();
        // (d) panel rows of D: inv(App) - I
        Dm[(p0 + rr) * D_W + cc] = sI[rr * 17 + cc] - ((rr == cc) ? 1.0f : 0.0f);
        __syncthreads();

        // (e) aug += D * Pold  — 128x256x16 rank-16 update, 16 tiles per wave (WMMA)
        for (int t = wave; t < 128; t += 8) {
            const int TI = t >> 4;
            const int TJ = t & 15;
            v8f acc;
            #pragma unroll
            for (int v = 0; v < 8; ++v)
                acc[v] = aug[(TI * 16 + v + 8 * hi) * AUG_W + TJ * 16 + n];
            #pragma unroll
            for (int k0 = 0; k0 < 16; k0 += 4) {
                v2f af, bf;
                af.x = Dm[(TI * 16 + n) * D_W + k0 + kb    ];
                af.y = Dm[(TI * 16 + n) * D_W + k0 + kb + 1];
                bf.x = Pold[(k0 + kb    ) * AUG_W + TJ * 16 + n];
                bf.y = Pold[(k0 + kb + 1) * AUG_W + TJ * 16 + n];
                acc = __builtin_amdgcn_wmma_f32_16x16x4_f32(
                        false, af, false, bf, (short)0, acc, false, false);
            }
            #pragma unroll
            for (int v = 0; v < 8; ++v)
                aug[(TI * 16 + v + 8 * hi) * AUG_W + TJ * 16 + n] = acc[v];
        }
        __syncthreads();
    }
    // aug[:, 128:256] now holds Kinv

    // ---------- Phase 3: 100 ADMM iterations, all operands LDS-resident ----------
    const float inv_rho = 1.0f / RHO;
    for (int it = 0; it < NITER; ++it) {
        if (tid < MTOT) vv[tid] = RHO * zv[tid] - yv[tid];
        __syncthreads();
        // rhs = SIGMA*x - q2 + A^T v   (column read of sA: odd stride -> conflict-free)
        if (tid < NX) {
            float s = SIGMA * xv[tid] - q2[tid];
            #pragma unroll 8
            for (int m = 0; m < MTOT; ++m) s += sA[m * A_W + tid] * vv[m];
            rhs[tid] = s;
        }
        __syncthreads();
        // x = Kinv * rhs   (row read of aug: odd stride -> conflict-free)
        if (tid < NX) {
            float s = 0.0f;
            #pragma unroll 8
            for (int j = 0; j < NX; ++j) s += aug[tid * AUG_W + NX + j] * rhs[j];
            xv[tid] = s;
        }
        __syncthreads();
        // Ax, z = min(Ax + y/rho, u), y += rho*(Ax - z)
        if (tid < MTOT) {
            float s = 0.0f;
            #pragma unroll 8
            for (int i = 0; i < NX; ++i) s += sA[tid * A_W + i] * xv[i];
            float zz = fminf(s + yv[tid] * inv_rho, uu[tid]);
            zv[tid] = zz;
            yv[tid] += RHO * (s - zz);
        }
        __syncthreads();
    }

    if (tid < NX) out[(size_t)b * NX + tid] = xv[tid];
}

extern "C" void kernel_launch(void* const* d_in, const int* in_sizes, int n_in,
                              void* d_out, int out_size, void* d_ws, size_t ws_size,
                              hipStream_t stream) {
    (void)n_in; (void)d_ws; (void)ws_size; (void)out_size;
    const float* x_in = (const float*)d_in[0];
    const float* Q    = (const float*)d_in[1];
    const float* q    = (const float*)d_in[2];
    const float* Ain  = (const float*)d_in[3];
    const float* bin  = (const float*)d_in[4];
    const float* Aeq  = (const float*)d_in[5];
    const float* beq  = (const float*)d_in[6];
    float* out = (float*)d_out;

    const int N = in_sizes[0] / NX;   // 1024

    // ~263 KB dynamic LDS per workgroup (within CDNA5's 320 KB/WGP) — opt in.
    hipFuncSetAttribute(reinterpret_cast<const void*>(admm_qp_kernel),
                        hipFuncAttributeMaxDynamicSharedMemorySize, SMEM_BYTES);

    admm_qp_kernel<<<dim3(N), dim3(256), SMEM_BYTES, stream>>>(
        x_in, Q, q, Ain, bin, Aeq, beq, out);
}